// yolov8_detection_target_11321533792584
// MI455X (gfx1250) — compile-verified
//
#include <hip/hip_runtime.h>

typedef __attribute__((ext_vector_type(2))) float v2f;
typedef __attribute__((ext_vector_type(4))) float v4f;
typedef __attribute__((ext_vector_type(8))) float v8f;

#define CONF 0.25f
#define NB2 512

// ---------------------------------------------------------------------------
// Pass 0: initialize cutoff (first row whose max-score < CONF; K if none).
// ---------------------------------------------------------------------------
__global__ void k_init(int* __restrict__ cutoff, int K) {
    if (threadIdx.x == 0 && blockIdx.x == 0) *cutoff = K;
}

// ---------------------------------------------------------------------------
// Pass 1: per-row max over 80 logits (streaming non-temporal b128 loads)
//         + box row-sums via V_WMMA_F32_16X16X4_F32 (B = all-ones).
// Each wave handles 32 consecutive rows: per-lane row max + two 16-row WMMAs.
// Row indices are CLAMPED (not branched) for loads so EXEC stays all-1s
// through the WMMA region; only stores/atomics are guarded. 32-bit indices
// keep addressing in SADDR + voffset (scale_offset) form.
// ---------------------------------------------------------------------------
__global__ __launch_bounds__(256)
void k_rowmax_boxsum(const float* __restrict__ logits,
                     const float* __restrict__ boxes,
                     float* __restrict__ scores,
                     float* __restrict__ boxsum,
                     int* __restrict__ cutoff,
                     int K)
{
    const int lane  = threadIdx.x & 31;
    const int gwave = blockIdx.x * (blockDim.x >> 5) + (threadIdx.x >> 5);
    const int base  = gwave * 32;
    const int row   = base + lane;
    const int rowc  = row < K ? row : K - 1;     // clamped: load-safe, no branch

    // ---- row-wise max over 80 fp32 logits: 20 x global_load_b128 (NT) ----
    {
        const v4f* lp = (const v4f*)logits;
        const int lidx = rowc * 20;              // 32-bit index, saddr+voffset form
        float m = -3.402823466e+38f;
        #pragma unroll
        for (int j = 0; j < 20; ++j) {
            v4f v = __builtin_nontemporal_load(lp + lidx + j);
            m = fmaxf(m, fmaxf(fmaxf(v.x, v.y), fmaxf(v.z, v.w)));
        }
        if (row < K) {
            scores[row] = m;
            // min over all failing rows == index of FIRST failing row
            if (m < CONF) atomicMin(cutoff, row);
        }
    }

    // ---- box row sums: D = A(16x4 boxes) x B(4x16 ones) + 0, f32 WMMA ----
    // A layout (32-bit A 16x4): lanes 0-15 hold M=lane,K={0,1};
    //                           lanes 16-31 hold M=lane-16,K={2,3}
    v2f bones; bones.x = 1.0f; bones.y = 1.0f;
    const int hi = lane >> 4;                    // 0: K-pair {0,1}, 1: {2,3}
    const int mr = lane & 15;                    // M index within tile

    int r0 = base + mr;           r0 = r0 < K ? r0 : K - 1;
    int r1 = base + 16 + mr;      r1 = r1 < K ? r1 : K - 1;
    const v2f* bp = (const v2f*)boxes;           // 2 coords per element
    v2f a0 = __builtin_nontemporal_load(bp + (r0 * 2 + hi));
    v2f a1 = __builtin_nontemporal_load(bp + (r1 * 2 + hi));

    v8f c0 = {};
    c0 = __builtin_amdgcn_wmma_f32_16x16x4_f32(
             false, a0, false, bones, (short)0, c0, false, false);
    v8f c1 = {};
    c1 = __builtin_amdgcn_wmma_f32_16x16x4_f32(
             false, a1, false, bones, (short)0, c1, false, false);

    // D layout (32-bit C/D 16x16): VGPR r = row M=r (lanes 0-15) / M=r+8
    // (lanes 16-31); B all-ones => all N columns identical. Lane 0 holds rows
    // base+0..7 (tile0) / base+16..23 (tile1); lane 16 holds base+8..15 /
    // base+24..31. One merged predicate, vector stores on the full-wave path.
    if ((lane & 15) == 0) {
        const int off = hi * 8;
        const int s0  = base + off;              // tile0 rows for this lane
        const int s1  = base + 16 + off;         // tile1 rows for this lane
        v4f lo0; lo0.x = c0[0]; lo0.y = c0[1]; lo0.z = c0[2]; lo0.w = c0[3];
        v4f hi0; hi0.x = c0[4]; hi0.y = c0[5]; hi0.z = c0[6]; hi0.w = c0[7];
        v4f lo1; lo1.x = c1[0]; lo1.y = c1[1]; lo1.z = c1[2]; lo1.w = c1[3];
        v4f hi1; hi1.x = c1[4]; hi1.y = c1[5]; hi1.z = c1[6]; hi1.w = c1[7];
        if (base + 32 <= K) {                    // full wave: 4 x b128 stores
            *(v4f*)(boxsum + s0)     = lo0;
            *(v4f*)(boxsum + s0 + 4) = hi0;
            *(v4f*)(boxsum + s1)     = lo1;
            *(v4f*)(boxsum + s1 + 4) = hi1;
        } else {                                 // tail wave: guarded scalars
            #pragma unroll
            for (int r = 0; r < 8; ++r) {
                if (s0 + r < K) boxsum[s0 + r] = c0[r];
                if (s1 + r < K) boxsum[s1 + r] = c1[r];
            }
        }
    }
}

// ---------------------------------------------------------------------------
// Pass 2: per-block partial sums of scores[i]+boxsum[i] for i < cutoff.
// Deterministic grid-stride + LDS tree reduction.
// ---------------------------------------------------------------------------
__global__ __launch_bounds__(256)
void k_partial(const float* __restrict__ scores,
               const float* __restrict__ boxsum,
               const int* __restrict__ cutoff,
               float* __restrict__ partials,
               int K)
{
    __shared__ float sm[256];
    const int m = *cutoff;
    float acc = 0.0f;
    for (int i = blockIdx.x * 256 + threadIdx.x; i < K; i += NB2 * 256) {
        if (i < m) acc += scores[i] + boxsum[i];
    }
    sm[threadIdx.x] = acc;
    __syncthreads();
    #pragma unroll
    for (int s = 128; s > 0; s >>= 1) {
        if (threadIdx.x < s) sm[threadIdx.x] += sm[threadIdx.x + s];
        __syncthreads();
    }
    if (threadIdx.x == 0) partials[blockIdx.x] = sm[0];
}

// ---------------------------------------------------------------------------
// Pass 3: final reduction of NB2 partials -> d_out[0].
// ---------------------------------------------------------------------------
__global__ __launch_bounds__(256)
void k_final(const float* __restrict__ partials, float* __restrict__ out)
{
    __shared__ float sm[256];
    float acc = 0.0f;
    for (int i = threadIdx.x; i < NB2; i += 256) acc += partials[i];
    sm[threadIdx.x] = acc;
    __syncthreads();
    #pragma unroll
    for (int s = 128; s > 0; s >>= 1) {
        if (threadIdx.x < s) sm[threadIdx.x] += sm[threadIdx.x + s];
        __syncthreads();
    }
    if (threadIdx.x == 0) out[0] = sm[0];
}

// ---------------------------------------------------------------------------
extern "C" void kernel_launch(void* const* d_in, const int* in_sizes, int n_in,
                              void* d_out, int out_size, void* d_ws, size_t ws_size,
                              hipStream_t stream)
{
    const float* logits = (const float*)d_in[0];
    const float* boxes  = (const float*)d_in[1];

    const int N = in_sizes[0] / 80;      // C = 80
    int K = N / 4;                       // RATIO = 0.25
    if (K < 1) K = 1;

    char*  ws       = (char*)d_ws;
    float* scores   = (float*)ws;                               // K floats
    float* boxsum   = (float*)(ws + (size_t)K * 4);             // K floats
    int*   cutoff   = (int*)  (ws + (size_t)K * 8);             // 1 int
    float* partials = (float*)(ws + (size_t)K * 8 + 16);        // NB2 floats

    k_init<<<1, 1, 0, stream>>>(cutoff, K);

    const int waves  = (K + 31) / 32;
    const int blocks = (waves + 7) / 8;  // 8 waves (256 threads) per block
    k_rowmax_boxsum<<<blocks, 256, 0, stream>>>(logits, boxes, scores, boxsum, cutoff, K);

    k_partial<<<NB2, 256, 0, stream>>>(scores, boxsum, cutoff, partials, K);
    k_final<<<1, 256, 0, stream>>>(partials, (float*)d_out);
}